// GCNEncoder_26414048871065
// MI455X (gfx1250) — compile-verified
//
#include <hip/hip_runtime.h>
#include <hip/hip_bf16.h>

typedef __attribute__((ext_vector_type(2))) float v2f;
typedef __attribute__((ext_vector_type(8))) float v8f;

// ---------------------------------------------------------------------------
// Degree / normalization kernels (GCN symmetric norm with self loops)
// ---------------------------------------------------------------------------
__global__ void deg_init_kernel(float* __restrict__ dinv, int n) {
    int i = blockIdx.x * blockDim.x + threadIdx.x;
    if (i < n) dinv[i] = 1.0f;                      // self loop contributes 1
}

__global__ void deg_count_kernel(const int* __restrict__ dst, float* __restrict__ dinv, int e) {
    int i = blockIdx.x * blockDim.x + threadIdx.x;
    if (i < e) {
        __hip_atomic_fetch_add(&dinv[dst[i]], 1.0f,
                               __ATOMIC_RELAXED, __HIP_MEMORY_SCOPE_AGENT);
    }
}

__global__ void deg_rsqrt_kernel(float* __restrict__ dinv, int n) {
    int i = blockIdx.x * blockDim.x + threadIdx.x;
    if (i < n) dinv[i] = rsqrtf(dinv[i]);           // deg >= 1 always
}

// ---------------------------------------------------------------------------
// WMMA fp32 GEMM:  H[N x Fout] = act(X)[N x Fin] * W[Fin x Fout]
// One wave computes a 16 x (16*COLS) slab of H with V_WMMA_F32_16X16X4_F32,
// reusing each A fragment across COLS accumulators (A traffic / COLS).
// A layout (16x4 f32): lane m = lane&15 holds row M=m, kgroup = lane>>4,
//   VGPR0 = K = 2*kgroup, VGPR1 = K = 2*kgroup+1.  B mirrors with N = lane&15.
// C/D layout (16x16 f32): VGPR v holds M = v + 8*(lane>>4), N = lane&15.
// N (=50000) is a multiple of 16; every Fout is a multiple of 16*COLS for the
// instantiation chosen by the host; only Fin=35 needs the K tail guard.
// ---------------------------------------------------------------------------
template <int COLS, bool RELU>
__global__ __launch_bounds__(256)
void gcn_gemm_wmma(const float* __restrict__ X, const float* __restrict__ W,
                   float* __restrict__ H, int Fin, int Fout,
                   int colGroups, int totalTiles) {
    int flat = blockIdx.x * blockDim.x + threadIdx.x;
    int wave = flat >> 5;                           // wave32
    int lane = flat & 31;
    if (wave >= totalTiles) return;                 // uniform per-wave exit

    int rowT = wave / colGroups;
    int colG = wave - rowT * colGroups;
    int row0 = rowT << 4;
    int col0 = colG * (COLS * 16);

    int m  = lane & 15;                             // M (for A) / N (for B)
    int kg = lane >> 4;                             // K group 0/1

    const float* __restrict__ xrow = X + (size_t)(row0 + m) * (size_t)Fin;
    const float* __restrict__ wcol = W + (size_t)(col0 + m);

    v8f c[COLS];
#pragma unroll
    for (int j = 0; j < COLS; ++j)
        c[j] = (v8f){0.f, 0.f, 0.f, 0.f, 0.f, 0.f, 0.f, 0.f};

    int kfull = Fin & ~15;
    int k0 = 0;
    for (; k0 < kfull; k0 += 16) {
#pragma unroll
        for (int s = 0; s < 4; ++s) {
            int ka = k0 + s * 4 + kg * 2;
            float ax = xrow[ka];
            float ay = xrow[ka + 1];
            if (RELU) { ax = fmaxf(ax, 0.f); ay = fmaxf(ay, 0.f); }
            v2f a = {ax, ay};
#pragma unroll
            for (int j = 0; j < COLS; ++j) {
                v2f b = {wcol[(size_t)ka * (size_t)Fout + j * 16],
                         wcol[(size_t)(ka + 1) * (size_t)Fout + j * 16]};
                c[j] = __builtin_amdgcn_wmma_f32_16x16x4_f32(
                           false, a, false, b, (short)0, c[j], false, false);
            }
        }
    }
    for (; k0 < Fin; k0 += 4) {                     // tail (Fin = 35 only)
        int ka = k0 + kg * 2;
        float ax = 0.f, ay = 0.f;
        bool g0 = (ka < Fin), g1 = (ka + 1 < Fin);
        if (g0) ax = xrow[ka];
        if (g1) ay = xrow[ka + 1];
        if (RELU) { ax = fmaxf(ax, 0.f); ay = fmaxf(ay, 0.f); }
        v2f a = {ax, ay};
#pragma unroll
        for (int j = 0; j < COLS; ++j) {
            v2f b = {0.f, 0.f};
            if (g0) b.x = wcol[(size_t)ka * (size_t)Fout + j * 16];
            if (g1) b.y = wcol[(size_t)(ka + 1) * (size_t)Fout + j * 16];
            c[j] = __builtin_amdgcn_wmma_f32_16x16x4_f32(
                       false, a, false, b, (short)0, c[j], false, false);
        }
    }

#pragma unroll
    for (int j = 0; j < COLS; ++j) {
        float* __restrict__ hout =
            H + (size_t)(row0 + kg * 8) * (size_t)Fout + (size_t)(col0 + j * 16 + m);
#pragma unroll
        for (int v = 0; v < 8; ++v)
            hout[(size_t)v * (size_t)Fout] = c[j][v];
    }
}

// ---------------------------------------------------------------------------
// OUT[i,f] = b[f] + H[i,f] * dinv[i]^2   (bias + self-loop term; also fully
// initializes the output buffer before the edge scatter adds into it)
// Fout is a power of two: f = t & fmask, i = t >> fshift.
// ---------------------------------------------------------------------------
__global__ void self_bias_kernel(const float* __restrict__ H,
                                 const float* __restrict__ dinv,
                                 const float* __restrict__ bias,
                                 float* __restrict__ O,
                                 int total, int fmask, int fshift) {
    int t = blockIdx.x * blockDim.x + threadIdx.x;
    if (t >= total) return;
    int f = t & fmask;
    int i = t >> fshift;
    float di = dinv[i];
    O[t] = bias[f] + H[t] * di * di;
}

// ---------------------------------------------------------------------------
// Edge scatter: for each edge e and each float4 feature chunk c:
//   OUT[dst[e], 4c..4c+3] += H[src[e], 4c..4c+3] * dinv[src]*dinv[dst]
// One thread per (edge, chunk): consecutive lanes read consecutive chunks of
// one H row (coalesced); src/dst/dinv reads are same-line broadcasts.
// ---------------------------------------------------------------------------
__global__ void edge_scatter_kernel(const float* __restrict__ H,
                                    const int* __restrict__ src,
                                    const int* __restrict__ dst,
                                    const float* __restrict__ dinv,
                                    float* __restrict__ O,
                                    int total, int cmask, int cshift, int Fout) {
    int t = blockIdx.x * blockDim.x + threadIdx.x;
    if (t >= total) return;
    int c = t & cmask;
    int e = t >> cshift;
    int s = src[e];
    int d = dst[e];
    float w = dinv[s] * dinv[d];
    const float4* h4 = reinterpret_cast<const float4*>(H + (size_t)s * (size_t)Fout);
    float4 h = h4[c];
    float* o = O + (size_t)d * (size_t)Fout + ((size_t)c << 2);
    __hip_atomic_fetch_add(o + 0, h.x * w, __ATOMIC_RELAXED, __HIP_MEMORY_SCOPE_AGENT);
    __hip_atomic_fetch_add(o + 1, h.y * w, __ATOMIC_RELAXED, __HIP_MEMORY_SCOPE_AGENT);
    __hip_atomic_fetch_add(o + 2, h.z * w, __ATOMIC_RELAXED, __HIP_MEMORY_SCOPE_AGENT);
    __hip_atomic_fetch_add(o + 3, h.w * w, __ATOMIC_RELAXED, __HIP_MEMORY_SCOPE_AGENT);
}

// ---------------------------------------------------------------------------
// Host-side orchestration
// ---------------------------------------------------------------------------
static int ilog2i(int v) { int r = 0; while ((1 << r) < v) ++r; return r; }

extern "C" void kernel_launch(void* const* d_in, const int* in_sizes, int n_in,
                              void* d_out, int out_size, void* d_ws, size_t ws_size,
                              hipStream_t stream) {
    const int FIN[6]  = {35, 64, 256, 512, 256, 64};
    const int FOUT[6] = {64, 256, 512, 256, 64, 16};

    const float* x   = (const float*)d_in[0];
    const int*   ei  = (const int*)d_in[1];
    const int Nn = in_sizes[0] / 35;        // 50000 (multiple of 16)
    const int E  = in_sizes[1] / 2;         // 800000
    const int* src = ei;                    // edge_index[0]
    const int* dst = ei + E;                // edge_index[1]

    // Workspace layout (floats): dinv[N] | H[N*512] | P[N*512] | Q[N*512]
    float* wsf  = (float*)d_ws;
    float* dinv = wsf;
    float* H    = wsf + (size_t)Nn;
    float* P    = H + (size_t)Nn * 512;
    float* Q    = P + (size_t)Nn * 512;

    const int TPB = 256;

    // --- degree / norm ---
    deg_init_kernel<<<(Nn + TPB - 1) / TPB, TPB, 0, stream>>>(dinv, Nn);
    deg_count_kernel<<<(E + TPB - 1) / TPB, TPB, 0, stream>>>(dst, dinv, E);
    deg_rsqrt_kernel<<<(Nn + TPB - 1) / TPB, TPB, 0, stream>>>(dinv, Nn);

    // --- 6 GCN layers ---
    const float* cur = x;
    for (int l = 0; l < 6; ++l) {
        const int fin  = FIN[l];
        const int fout = FOUT[l];
        const float* W = (const float*)d_in[2 + 2 * l];
        const float* b = (const float*)d_in[3 + 2 * l];
        float* out = (l == 5) ? (float*)d_out : ((l & 1) ? Q : P);

        // GEMM: H = act(cur) @ W
        const int rowTiles = Nn / 16;
        const int cols = (fout % 64 == 0) ? 4 : 1;  // 16x64 slab when possible
        const int colGroups = fout / (16 * cols);
        const int totalTiles = rowTiles * colGroups;
        const int gemmThreads = totalTiles * 32;
        const int gemmBlocks = (gemmThreads + TPB - 1) / TPB;
        if (cols == 4) {
            if (l > 0)
                gcn_gemm_wmma<4, true><<<gemmBlocks, TPB, 0, stream>>>(
                    cur, W, H, fin, fout, colGroups, totalTiles);
            else
                gcn_gemm_wmma<4, false><<<gemmBlocks, TPB, 0, stream>>>(
                    cur, W, H, fin, fout, colGroups, totalTiles);
        } else {
            if (l > 0)
                gcn_gemm_wmma<1, true><<<gemmBlocks, TPB, 0, stream>>>(
                    cur, W, H, fin, fout, colGroups, totalTiles);
            else
                gcn_gemm_wmma<1, false><<<gemmBlocks, TPB, 0, stream>>>(
                    cur, W, H, fin, fout, colGroups, totalTiles);
        }

        // out = b + H * dinv^2   (self loop + bias, full init of out)
        const int totalSB = Nn * fout;
        self_bias_kernel<<<(totalSB + TPB - 1) / TPB, TPB, 0, stream>>>(
            H, dinv, b, out, totalSB, fout - 1, ilog2i(fout));

        // out[dst] += H[src] * norm   over all edges
        const int chunks = fout / 4;
        const int totalSC = E * chunks;
        edge_scatter_kernel<<<(totalSC + TPB - 1) / TPB, TPB, 0, stream>>>(
            H, src, dst, dinv, out, totalSC, chunks - 1, ilog2i(chunks), fout);

        cur = out;
    }
}